// P2M2_CDR_36146444763708
// MI455X (gfx1250) — compile-verified
//
#include <hip/hip_runtime.h>
#include <stdint.h>

#define UCNT 150000
#define ICNT 100000
#define NNODE (UCNT + ICNT)   // 250000
#define DIM 64
#define ECNT 1500000
#define NLAYER 3

// ---------------- degree counting (int atomics, deterministic counts) --------
__global__ void k_count(const int* __restrict__ eu, const int* __restrict__ ei,
                        int* __restrict__ cnt_u, int* __restrict__ cnt_i) {
    int e = blockIdx.x * 256 + threadIdx.x;
    if (e >= ECNT) return;
    atomicAdd(&cnt_u[eu[e]], 1);
    atomicAdd(&cnt_i[ei[e]], 1);
}

// ---------------- exclusive scan: pass1 (per-1024-chunk scan + chunk totals) -
__global__ void k_scan1(const int* __restrict__ cnt, int n,
                        int* __restrict__ out, int* __restrict__ sums) {
    __shared__ int lds[256];
    int t = threadIdx.x;
    int base = blockIdx.x * 1024 + t * 4;
    int x0 = (base + 0 < n) ? cnt[base + 0] : 0;
    int x1 = (base + 1 < n) ? cnt[base + 1] : 0;
    int x2 = (base + 2 < n) ? cnt[base + 2] : 0;
    int x3 = (base + 3 < n) ? cnt[base + 3] : 0;
    lds[t] = x0 + x1 + x2 + x3;
    __syncthreads();
    for (int o = 1; o < 256; o <<= 1) {
        int v = lds[t];
        int u = (t >= o) ? lds[t - o] : 0;
        __syncthreads();
        lds[t] = v + u;
        __syncthreads();
    }
    if (t == 255) sums[blockIdx.x] = lds[255];
    int e0 = (t == 0) ? 0 : lds[t - 1];
    int e1 = e0 + x0, e2 = e1 + x1, e3 = e2 + x2;
    if (base + 0 < n) out[base + 0] = e0;
    if (base + 1 < n) out[base + 1] = e1;
    if (base + 2 < n) out[base + 2] = e2;
    if (base + 3 < n) out[base + 3] = e3;
}

// pass2: single block exclusive-scans the chunk totals (nb <= 256)
__global__ void k_scan2(int* __restrict__ sums, int nb) {
    __shared__ int lds[256];
    int t = threadIdx.x;
    int x = (t < nb) ? sums[t] : 0;
    lds[t] = x;
    __syncthreads();
    for (int o = 1; o < 256; o <<= 1) {
        int v = lds[t];
        int u = (t >= o) ? lds[t - o] : 0;
        __syncthreads();
        lds[t] = v + u;
        __syncthreads();
    }
    if (t < nb) sums[t] = (t == 0) ? 0 : lds[t - 1];
}

// pass3: add chunk offsets; write terminal offset off[n] = total
__global__ void k_scan3(int* __restrict__ off, const int* __restrict__ sums,
                        int n, int total) {
    int i = blockIdx.x * 256 + threadIdx.x;
    if (i < n) off[i] += sums[i >> 10];
    if (i == 0) off[n] = total;
}

// ---------------- d^{-1/2} --------------------------------------------------
__global__ void k_dinv(const int* __restrict__ cnt_u, const int* __restrict__ cnt_i,
                       float* __restrict__ dinv) {
    int n = blockIdx.x * 256 + threadIdx.x;
    if (n >= NNODE) return;
    int c = (n < UCNT) ? cnt_u[n] : cnt_i[n - UCNT];
    dinv[n] = 1.0f / sqrtf((float)c + 1e-7f);
}

// ---------------- CSR fill: (neighbor, weight) packed as int2 ---------------
__global__ void k_fill(const int* __restrict__ eu, const int* __restrict__ ei,
                       const float* __restrict__ dinv,
                       int* __restrict__ cur_u, int* __restrict__ cur_i,
                       int2* __restrict__ csr_u, int2* __restrict__ csr_i) {
    int e = blockIdx.x * 256 + threadIdx.x;
    if (e >= ECNT) return;
    int u = eu[e], i = ei[e];
    float w = dinv[u] * dinv[UCNT + i];
    int wb = __float_as_int(w);
    int pu = atomicAdd(&cur_u[u], 1);
    csr_u[pu] = make_int2(i, wb);
    int pi = atomicAdd(&cur_i[i], 1);
    csr_i[pi] = make_int2(u, wb);
}

// ---------------- init: h0 = ego = [u_emb; v_emb], acc = ego (b128 traffic) -
__global__ void k_init(const float4* __restrict__ u4, const float4* __restrict__ v4,
                       float4* __restrict__ h, float4* __restrict__ acc) {
    int idx = blockIdx.x * 256 + threadIdx.x;       // NNODE*16 = 4,000,000 exact
    float4 val = (idx < UCNT * 16) ? u4[idx] : v4[idx - UCNT * 16];
    h[idx] = val;
    acc[idx] = val;
}

// ---------------- one propagation layer: pull-based SpMM --------------------
// One wave32 per destination node; lane owns float2 columns [2*lane, 2*lane+1].
// Adjacency chunk (up to 32 (col,w) pairs) is staged to wave-private LDS with
// the gfx1250 async global->LDS path (ASYNCcnt), then broadcast-read.
__global__ __launch_bounds__(256) void k_spmv(
    const float* __restrict__ hcur, float* __restrict__ hnext,
    float* __restrict__ acc,
    const int* __restrict__ off_u, const int* __restrict__ off_i,
    const int2* __restrict__ csr_u, const int2* __restrict__ csr_i) {
    __shared__ int2 stage[8][32];                   // 8 waves/block, 2 KB
    int lane = threadIdx.x & 31;
    int wave = threadIdx.x >> 5;
    int n = blockIdx.x * 8 + wave;                  // grid = NNODE/8, exact

    bool isU = (n < UCNT);
    const int*  off = isU ? off_u : off_i;
    const int2* csr = isU ? csr_u : csr_i;
    int row     = isU ? n : (n - UCNT);
    int srcBase = isU ? UCNT : 0;                   // users gather item rows & v.v.

    int s = off[row];
    int e = off[row + 1];
    float ax = 0.0f, ay = 0.0f;

    // raw 32-bit LDS byte address of this lane's staging slot (flat aperture:
    // low 32 bits of a generic LDS pointer are the workgroup-relative offset)
    unsigned ldsAddr = (unsigned)(uintptr_t)(&stage[wave][lane]);

    for (int k = s; k < e; k += 32) {
        int m = e - k;
        if (m > 32) m = 32;
        // clamp tail lanes to a valid entry; their slots are simply not consumed
        uint64_t gaddr = (uint64_t)(csr + k + ((lane < m) ? lane : (m - 1)));
        // gfx1250 async copy: 8 bytes/lane, global -> LDS, tracked by ASYNCcnt
        asm volatile("global_load_async_to_lds_b64 %0, %1, off"
                     :: "v"(ldsAddr), "v"(gaddr) : "memory");
        asm volatile("s_wait_asynccnt 0x0" ::: "memory");

        for (int t = 0; t < m; ++t) {
            int2 pr = stage[wave][t];               // LDS broadcast read
            float wgt = __int_as_float(pr.y);
            const float2 hv =
                ((const float2*)hcur)[(size_t)(srcBase + pr.x) * 32 + lane];
            ax = fmaf(wgt, hv.x, ax);
            ay = fmaf(wgt, hv.y, ay);
        }
    }

    size_t o = (size_t)n * 32 + lane;
    ((float2*)hnext)[o] = make_float2(ax, ay);
    float2 a = ((float2*)acc)[o];                   // running layer sum
    a.x += ax; a.y += ay;
    ((float2*)acc)[o] = a;
}

// ---------------- final: out = acc/4 ; item block += v_emb ------------------
__global__ void k_final(float4* __restrict__ out, const float4* __restrict__ v4) {
    int idx = blockIdx.x * 256 + threadIdx.x;       // 4,000,000 exact
    float4 o = out[idx];
    o.x *= 0.25f; o.y *= 0.25f; o.z *= 0.25f; o.w *= 0.25f;
    if (idx >= UCNT * 16) {
        float4 v = v4[idx - UCNT * 16];
        o.x += v.x; o.y += v.y; o.z += v.z; o.w += v.w;
    }
    out[idx] = o;
}

// ---------------------------------------------------------------------------
extern "C" void kernel_launch(void* const* d_in, const int* in_sizes, int n_in,
                              void* d_out, int out_size, void* d_ws, size_t ws_size,
                              hipStream_t stream) {
    (void)in_sizes; (void)n_in; (void)out_size; (void)ws_size;
    const float* u_emb = (const float*)d_in[0];
    const float* v_emb = (const float*)d_in[1];
    const int*   eu    = (const int*)d_in[2];
    const int*   ei    = (const int*)d_in[3];

    // workspace carve-out (~150 MB total)
    char* p = (char*)d_ws;
    auto alloc = [&](size_t b) { void* r = (void*)p; p += (b + 255) & ~(size_t)255; return r; };
    int*   cnt_u  = (int*)alloc((size_t)UCNT * 4);
    int*   cnt_i  = (int*)alloc((size_t)ICNT * 4);
    int*   off_u  = (int*)alloc((size_t)(UCNT + 1) * 4);
    int*   off_i  = (int*)alloc((size_t)(ICNT + 1) * 4);
    int*   cur_u  = (int*)alloc((size_t)UCNT * 4);
    int*   cur_i  = (int*)alloc((size_t)ICNT * 4);
    int*   sums_u = (int*)alloc(256 * 4);
    int*   sums_i = (int*)alloc(256 * 4);
    float* dinv   = (float*)alloc((size_t)NNODE * 4);
    int2*  csr_u  = (int2*)alloc((size_t)ECNT * 8);
    int2*  csr_i  = (int2*)alloc((size_t)ECNT * 8);
    float* h0     = (float*)alloc((size_t)NNODE * DIM * 4);
    float* h1     = (float*)alloc((size_t)NNODE * DIM * 4);

    hipMemsetAsync(cnt_u, 0, (size_t)UCNT * 4, stream);
    hipMemsetAsync(cnt_i, 0, (size_t)ICNT * 4, stream);

    k_count<<<(ECNT + 255) / 256, 256, 0, stream>>>(eu, ei, cnt_u, cnt_i);

    int nbU = (UCNT + 1023) / 1024;   // 147 (<= 256)
    int nbI = (ICNT + 1023) / 1024;   // 98  (<= 256)
    k_scan1<<<nbU, 256, 0, stream>>>(cnt_u, UCNT, off_u, sums_u);
    k_scan2<<<1, 256, 0, stream>>>(sums_u, nbU);
    k_scan3<<<(UCNT + 255) / 256, 256, 0, stream>>>(off_u, sums_u, UCNT, ECNT);
    k_scan1<<<nbI, 256, 0, stream>>>(cnt_i, ICNT, off_i, sums_i);
    k_scan2<<<1, 256, 0, stream>>>(sums_i, nbI);
    k_scan3<<<(ICNT + 255) / 256, 256, 0, stream>>>(off_i, sums_i, ICNT, ECNT);

    k_dinv<<<(NNODE + 255) / 256, 256, 0, stream>>>(cnt_u, cnt_i, dinv);

    hipMemcpyAsync(cur_u, off_u, (size_t)UCNT * 4, hipMemcpyDeviceToDevice, stream);
    hipMemcpyAsync(cur_i, off_i, (size_t)ICNT * 4, hipMemcpyDeviceToDevice, stream);

    k_fill<<<(ECNT + 255) / 256, 256, 0, stream>>>(eu, ei, dinv, cur_u, cur_i, csr_u, csr_i);

    float* acc = (float*)d_out;
    k_init<<<NNODE * 16 / 256, 256, 0, stream>>>((const float4*)u_emb,
                                                 (const float4*)v_emb,
                                                 (float4*)h0, (float4*)acc);

    const int spmvGrid = NNODE / 8;   // 31250, exact
    k_spmv<<<spmvGrid, 256, 0, stream>>>(h0, h1, acc, off_u, off_i, csr_u, csr_i);
    k_spmv<<<spmvGrid, 256, 0, stream>>>(h1, h0, acc, off_u, off_i, csr_u, csr_i);
    k_spmv<<<spmvGrid, 256, 0, stream>>>(h0, h1, acc, off_u, off_i, csr_u, csr_i);

    k_final<<<NNODE * 16 / 256, 256, 0, stream>>>((float4*)d_out, (const float4*)v_emb);
}